// CameraViewTransformerLSSVoxel_18133351923973
// MI455X (gfx1250) — compile-verified
//
#include <hip/hip_runtime.h>
#include <hip/hip_bf16.h>

// ---------------------------------------------------------------------------
// LSS camera->BEV voxel transform for MI455X (gfx1250, wave32, WMMA).
// Stage 1: frustum scatter with global_atomic_add_f32 (L2-resident 42MB acc).
// Stage 2: 128x640x16384 fp32 GEMM via v_wmma_f32_16x16x4_f32, LDS-staged B.
// Stage 3: per-channel BN stats + fused scale/shift/ReLU.
// ---------------------------------------------------------------------------

#define NCAM   6
#define DBIN   48
#define HFEAT  64
#define WFEAT  176
#define CCTX   80
#define BEVH   128
#define BEVW   128
#define BEVZ   8
#define BEVC   128
#define KDIM   640        // CCTX * BEVZ
#define NPIX   16384      // BEVH * BEVW
#define NVOX   131072     // BEVZ * BEVH * BEVW
#define LDSP   642        // padded LDS pitch (642 % 64 == 2 -> conflict-free)

typedef float v2f __attribute__((ext_vector_type(2)));
typedef float v8f __attribute__((ext_vector_type(8)));

// ---------------------------------------------------------------- zero scratch
__global__ void lss_zero(float* __restrict__ p, int n) {
  int i = blockIdx.x * blockDim.x + threadIdx.x;
  int stride = gridDim.x * blockDim.x;
  for (; i < n; i += stride) p[i] = 0.0f;
}

// ------------------------------------------------------------- frustum scatter
// one wave32 per feature pixel; lanes split the 80 context channels.
__global__ void __launch_bounds__(256)
lss_scatter(const float* __restrict__ dp, const float* __restrict__ ctx,
            const float* __restrict__ intr, const float* __restrict__ c2e,
            float* __restrict__ vox, float* __restrict__ cnt) {
  const int lane = threadIdx.x & 31;
  const int wave = threadIdx.x >> 5;
  const int pix  = blockIdx.x * 8 + wave;
  if (pix >= NCAM * HFEAT * WFEAT) return;
  const int n = pix / (HFEAT * WFEAT);
  const int r = pix - n * (HFEAT * WFEAT);
  const int h = r / WFEAT;
  const int w = r - h * WFEAT;

  const float fx = intr[n * 9 + 0], cx = intr[n * 9 + 2];
  const float fy = intr[n * 9 + 4], cy = intr[n * 9 + 5];
  const float R00 = c2e[n*16+0], R01 = c2e[n*16+1], R02 = c2e[n*16+2],  t0 = c2e[n*16+3];
  const float R10 = c2e[n*16+4], R11 = c2e[n*16+5], R12 = c2e[n*16+6],  t1 = c2e[n*16+7];
  const float R20 = c2e[n*16+8], R21 = c2e[n*16+9], R22 = c2e[n*16+10], t2 = c2e[n*16+11];

  // per-pixel context row, held in registers across all 48 depth bins
  const int cbase = (n * CCTX) * HFEAT * WFEAT + h * WFEAT + w;
  const float f0 = ctx[cbase + (lane      ) * (HFEAT * WFEAT)];
  const float f1 = ctx[cbase + (lane + 32 ) * (HFEAT * WFEAT)];
  const float f2 = (lane < 16) ? ctx[cbase + (lane + 64) * (HFEAT * WFEAT)] : 0.0f;

  const float u  = ((float)w + 0.5f) * 4.0f;
  const float v  = ((float)h + 0.5f) * 4.0f;
  const float xr = (u - cx) / fx;
  const float yr = (v - cy) / fy;
  const int dpbase = (n * DBIN) * HFEAT * WFEAT + h * WFEAT + w;

  for (int d = 0; d < DBIN; ++d) {
    const float Z  = 1.0f + (float)d * (59.0f / 47.0f);   // linspace(1,60,48)
    const float Xc = xr * Z, Yc = yr * Z;
    const float ex = R00 * Xc + R01 * Yc + R02 * Z + t0;
    const float ey = R10 * Xc + R11 * Yc + R12 * Z + t1;
    const float ez = R20 * Xc + R21 * Yc + R22 * Z + t2;
    const int ix = (int)floorf((ex + 50.0f) * 1.28f);     // / (100/128)
    const int iy = (int)floorf((ey + 50.0f) * 1.28f);
    const int iz = (int)floorf(ez + 5.0f);                // / 1.0
    if (ix < 0 || ix >= BEVW || iy < 0 || iy >= BEVH || iz < 0 || iz >= BEVZ)
      continue;                                           // wave-uniform branch
    const int vidx = (iz * BEVH + iy) * BEVW + ix;
    const float dpv = dp[dpbase + d * (HFEAT * WFEAT)];
    float* dst = vox + (size_t)vidx * CCTX;               // 80 contiguous floats
    atomicAdd(dst + lane,        dpv * f0);
    atomicAdd(dst + 32 + lane,   dpv * f1);
    if (lane < 16) atomicAdd(dst + 64 + lane, dpv * f2);
    if (lane == 0) atomicAdd(cnt + vidx, 1.0f);
  }
}

// -------------------------------------------------- WMMA GEMM: y = W * bev_in
// WG = 256 threads (8 waves); WG owns all 128 out-channels x 16 pixels.
__global__ void __launch_bounds__(256)
lss_gemm(const float* __restrict__ cw, const float* __restrict__ vox,
         const float* __restrict__ cnt, float* __restrict__ y) {
  __shared__ float Bs[16 * LDSP];                         // B tile, transposed
  const int tid  = threadIdx.x;
  const int pix0 = blockIdx.x * 16;

  // stage + normalize B tile: Bs[n][k] = vox[vidx(k,pix)][c] / max(cnt,1)
  #pragma unroll
  for (int it = 0; it < (KDIM * 16) / 256; ++it) {
    const int idx  = it * 256 + tid;
    const int k    = idx >> 4;                            // 0..639  (= c*8+z)
    const int ncol = idx & 15;
    const int c    = k >> 3;
    const int z    = k & 7;
    const int pix  = pix0 + ncol;
    const int hh   = pix >> 7, ww = pix & 127;
    const int vidx = (z * BEVH + hh) * BEVW + ww;
    const float val = vox[(size_t)vidx * CCTX + c];
    const float cv  = cnt[vidx];
    Bs[ncol * LDSP + k] = val / fmaxf(cv, 1.0f);
  }
  __syncthreads();

  const int lane = tid & 31;
  const int lm   = lane & 15;        // row-in-tile (A: M, B: N)
  const int hi   = lane >> 4;        // selects K pair {0,1} vs {2,3}
  const int m0   = (tid >> 5) * 16;  // this wave's 16 output channels
  const float* arow = cw + (size_t)(m0 + lm) * KDIM;      // W row, contiguous K
  const float* brow = Bs + lm * LDSP;                     // B col, contiguous K

  v8f acc = {0.f, 0.f, 0.f, 0.f, 0.f, 0.f, 0.f, 0.f};
  #pragma unroll 4
  for (int k0 = 0; k0 < KDIM; k0 += 4) {
    v2f a = *(const v2f*)(arow + k0 + 2 * hi);            // global_load_b64
    v2f b = *(const v2f*)(brow + k0 + 2 * hi);            // ds_load_b64
    acc = __builtin_amdgcn_wmma_f32_16x16x4_f32(
        /*neg_a=*/false, a, /*neg_b=*/false, b,
        /*c_mod=*/(short)0, acc, /*reuse_a=*/false, /*reuse_b=*/false);
  }

  // D layout (ISA 7.12.2): VGPR i -> M = i + 8*hi, N = lm
  #pragma unroll
  for (int i = 0; i < 8; ++i) {
    const int m = m0 + hi * 8 + i;
    y[(size_t)m * NPIX + pix0 + lm] = acc[i];
  }
}

// -------------------------------------------------------- BN stats per channel
__global__ void __launch_bounds__(256)
lss_bnstats(const float* __restrict__ y, const float* __restrict__ gamma,
            const float* __restrict__ beta, float* __restrict__ scale,
            float* __restrict__ shift) {
  __shared__ float s1[256];
  __shared__ float s2[256];
  const int m = blockIdx.x, tid = threadIdx.x;
  const float* ym = y + (size_t)m * NPIX;
  float s = 0.0f, q = 0.0f;
  for (int i = tid; i < NPIX; i += 256) {
    const float v = ym[i];
    s += v;
    q += v * v;
  }
  s1[tid] = s; s2[tid] = q;
  __syncthreads();
  for (int off = 128; off > 0; off >>= 1) {
    if (tid < off) { s1[tid] += s1[tid + off]; s2[tid] += s2[tid + off]; }
    __syncthreads();
  }
  if (tid == 0) {
    const float mean = s1[0] * (1.0f / NPIX);
    const float var  = s2[0] * (1.0f / NPIX) - mean * mean;
    const float sc   = gamma[m] * rsqrtf(var + 1e-5f);
    scale[m] = sc;
    shift[m] = beta[m] - mean * sc;
  }
}

// ------------------------------------------------------- BN apply + ReLU + out
__global__ void lss_bnrelu(const float* __restrict__ y,
                           const float* __restrict__ scale,
                           const float* __restrict__ shift,
                           float* __restrict__ out, int n) {
  int i = blockIdx.x * blockDim.x + threadIdx.x;
  int stride = gridDim.x * blockDim.x;
  for (; i < n; i += stride) {
    const int m = i >> 14;                                // i / NPIX
    out[i] = fmaxf(y[i] * scale[m] + shift[m], 0.0f);
  }
}

// ---------------------------------------------------------------------------
extern "C" void kernel_launch(void* const* d_in, const int* in_sizes, int n_in,
                              void* d_out, int out_size, void* d_ws, size_t ws_size,
                              hipStream_t stream) {
  const float* dp    = (const float*)d_in[0];  // depth_prob (1,6,48,64,176)
  const float* ctx   = (const float*)d_in[1];  // context    (1,6,80,64,176)
  const float* intr  = (const float*)d_in[2];  // intrinsics (1,6,3,3)
  const float* c2e   = (const float*)d_in[3];  // cam2ego    (1,6,4,4)
  const float* cw    = (const float*)d_in[4];  // conv_w     (128,640)
  const float* gamma = (const float*)d_in[5];  // (128,)
  const float* beta  = (const float*)d_in[6];  // (128,)
  float* out = (float*)d_out;                  // (1,128,128,128)

  // workspace layout (floats): vox | cnt | y | scale | shift  (~51 MB total)
  float* ws    = (float*)d_ws;
  float* vox   = ws;                                   // NVOX*CCTX = 10,485,760
  float* cnt   = vox + (size_t)NVOX * CCTX;            // NVOX      =    131,072
  float* ybuf  = cnt + NVOX;                           // BEVC*NPIX =  2,097,152
  float* scale = ybuf + (size_t)BEVC * NPIX;           // 128
  float* shift = scale + BEVC;                         // 128

  const int accum_floats = NVOX * CCTX + NVOX;
  lss_zero<<<2048, 256, 0, stream>>>(vox, accum_floats);

  const int npixels = NCAM * HFEAT * WFEAT;            // 67,584 (8 waves/block)
  lss_scatter<<<npixels / 8, 256, 0, stream>>>(dp, ctx, intr, c2e, vox, cnt);

  lss_gemm<<<NPIX / 16, 256, 0, stream>>>(cw, vox, cnt, ybuf);

  lss_bnstats<<<BEVC, 256, 0, stream>>>(ybuf, gamma, beta, scale, shift);

  lss_bnrelu<<<2048, 256, 0, stream>>>(ybuf, scale, shift, out, BEVC * NPIX);
}